// GCN_8710193677018
// MI455X (gfx1250) — compile-verified
//
#include <hip/hip_runtime.h>
#include <stdint.h>

#define N_NODES 100000
#define N_EDGES 1000000
#define D 64

typedef __attribute__((ext_vector_type(8))) float v8f;
typedef __attribute__((ext_vector_type(2))) float v2f;

// ---------------------------------------------------------------------------
// Phase 0: zero the aggregation scratch with b128 stores.
//   agg: N_NODES*D floats = 1,600,000 float4
//   deg: N_NODES floats   =    25,000 float4
// grid covers agg exactly (1.6M/64 threads... 1,600,000 float4 / 256 = 6250).
// ---------------------------------------------------------------------------
__global__ void __launch_bounds__(256)
init_ws(float4* __restrict__ agg4, float4* __restrict__ deg4) {
    const int i = blockIdx.x * 256 + threadIdx.x;                // 0 .. 1.6M-1
    agg4[i] = make_float4(0.f, 0.f, 0.f, 0.f);
    if (i < N_NODES / 4)
        deg4[i] = make_float4(0.f, 0.f, 0.f, 0.f);
}

// ---------------------------------------------------------------------------
// Phase 1: edge scatter.  16 lanes per edge; each lane moves 4 channels:
// one global_load_b128 gather from x[src], four relaxed agent-scope f32
// atomic adds into agg[dst] (no-return global_atomic_add_f32, L2-resident
// target).  N_EDGES*16 threads = 62500 blocks of 256 exactly -> no tail.
// ---------------------------------------------------------------------------
__global__ void __launch_bounds__(256)
scatter_accum(const float* __restrict__ x,
              const long long* __restrict__ ei,   // [2, N_EDGES] int64
              float* __restrict__ agg,
              float* __restrict__ deg) {
    const int t    = blockIdx.x * 256 + threadIdx.x;
    const int e    = t >> 4;           // edge id (16 lanes per edge)
    const int c    = (t & 15) * 4;     // channel base for this lane

    const int src = (int)ei[e];             // row 0: message sources
    const int dst = (int)ei[N_EDGES + e];   // row 1: aggregation targets

    const float4 v = *(const float4*)(x + (long long)src * D + c);
    float* a = agg + (long long)dst * D + c;
    __hip_atomic_fetch_add(a + 0, v.x, __ATOMIC_RELAXED, __HIP_MEMORY_SCOPE_AGENT);
    __hip_atomic_fetch_add(a + 1, v.y, __ATOMIC_RELAXED, __HIP_MEMORY_SCOPE_AGENT);
    __hip_atomic_fetch_add(a + 2, v.z, __ATOMIC_RELAXED, __HIP_MEMORY_SCOPE_AGENT);
    __hip_atomic_fetch_add(a + 3, v.w, __ATOMIC_RELAXED, __HIP_MEMORY_SCOPE_AGENT);
    if ((t & 15) == 0)
        __hip_atomic_fetch_add(deg + dst, 1.0f, __ATOMIC_RELAXED, __HIP_MEMORY_SCOPE_AGENT);
}

// ---------------------------------------------------------------------------
// Phase 2: out = relu( (agg/deg) @ Wl^T + bl + x @ Wr^T )
// 128 threads = 4 waves per block; block owns a 16-row tile, wave w owns
// column tile [16w, 16w+16).  K=64 per matrix -> 16 v_wmma_f32_16x16x4_f32
// steps each, 32 total, accumulating into one v8f (full f32 precision).
//
// Lane layouts per CDNA5 ISA 7.12.2 (wave32):
//   A 16x4 f32 : lane l -> row (l&15); VGPR0/1 hold K = (l>>4)*2 + {0,1}
//   B 4x16 f32 : lane l -> col (l&15); VGPR0/1 hold K = (l>>4)*2 + {0,1}
//   C/D 16x16  : VGPR v, lane l -> row v + (l>>4)*8, col (l&15)
// 100000 % 16 == 0 -> no tail, EXEC stays all-ones (WMMA requirement).
// ---------------------------------------------------------------------------
__global__ void __launch_bounds__(128)
sage_gemm(const float* __restrict__ x,
          const float* __restrict__ agg,
          const float* __restrict__ deg,
          const float* __restrict__ Wl,   // [D_OUT][D_IN] row-major
          const float* __restrict__ bl,   // [D_OUT]
          const float* __restrict__ Wr,   // [D_OUT][D_IN] row-major
          float* __restrict__ out) {
    const int lane = threadIdx.x & 31;
    const int wave = threadIdx.x >> 5;          // 0..3 -> column tile
    const int half = lane >> 4;                 // 0 or 1
    const int l15  = lane & 15;
    const int row0 = blockIdx.x * 16;
    const int n    = wave * 16 + l15;           // output column (B/D lanes)
    const int arow = row0 + l15;                // A-matrix row for this lane

    const float rdeg = 1.0f / fmaxf(deg[arow], 1.0f);

    const float* aAgg = agg + (long long)arow * D;
    const float* aX   = x   + (long long)arow * D;
    const float* bWl  = Wl  + (long long)n * D;
    const float* bWr  = Wr  + (long long)n * D;

    v8f acc = {};

    // mean @ Wl^T  (mean computed on the fly: agg * 1/max(deg,1))
    #pragma unroll
    for (int ks = 0; ks < 16; ++ks) {
        const int kb = ks * 4 + half * 2;
        v2f a, b;
        a.x = aAgg[kb]     * rdeg;
        a.y = aAgg[kb + 1] * rdeg;
        b.x = bWl[kb];
        b.y = bWl[kb + 1];
        acc = __builtin_amdgcn_wmma_f32_16x16x4_f32(
                  false, a, false, b, (short)0, acc, false, false);
    }
    // x @ Wr^T
    #pragma unroll
    for (int ks = 0; ks < 16; ++ks) {
        const int kb = ks * 4 + half * 2;
        v2f a, b;
        a.x = aX[kb];
        a.y = aX[kb + 1];
        b.x = bWr[kb];
        b.y = bWr[kb + 1];
        acc = __builtin_amdgcn_wmma_f32_16x16x4_f32(
                  false, a, false, b, (short)0, acc, false, false);
    }

    const float bias = bl[n];
    #pragma unroll
    for (int v = 0; v < 8; ++v) {
        const int m = half * 8 + v;             // D-matrix row for VGPR v
        const float r = acc[v] + bias;
        out[(long long)(row0 + m) * D + n] = fmaxf(r, 0.0f);
    }
}

// ---------------------------------------------------------------------------
extern "C" void kernel_launch(void* const* d_in, const int* in_sizes, int n_in,
                              void* d_out, int out_size, void* d_ws, size_t ws_size,
                              hipStream_t stream) {
    const float*     x  = (const float*)d_in[0];
    const long long* ei = (const long long*)d_in[1];   // int64 per reference
    const float*     Wl = (const float*)d_in[2];
    const float*     bl = (const float*)d_in[3];
    const float*     Wr = (const float*)d_in[4];
    float* out = (float*)d_out;

    float* agg = (float*)d_ws;                          // N_NODES * D floats
    float* deg = agg + (size_t)N_NODES * D;             // N_NODES floats

    // Phase 0: zero agg + deg (b128 stores; 1.6M float4 / 256 = 6250 blocks)
    init_ws<<<(N_NODES * D / 4) / 256, 256, 0, stream>>>(
        (float4*)agg, (float4*)deg);

    // Phase 1: scatter (16 lanes/edge; 16M threads = 62500 blocks exactly)
    scatter_accum<<<(N_EDGES * 16) / 256, 256, 0, stream>>>(x, ei, agg, deg);

    // Phase 2: WMMA GEMM + bias + ReLU (6250 row-tiles)
    sage_gemm<<<N_NODES / 16, 128, 0, stream>>>(x, agg, deg, Wl, bl, Wr, out);
}